// SR_GNN_27169963115103
// MI455X (gfx1250) — compile-verified
//
#include <hip/hip_runtime.h>
#include <hip/hip_bf16.h>

#define N_NODES   100000
#define N_EDGES   200000
#define N_GRAPHS  4096
#define EMB       64
#define HID       100
#define IN_DIM    321
#define NUM_ITEMS 50000
#define NTILES_GATE 19   // ceil(300/16)

typedef float v2f __attribute__((ext_vector_type(2)));
typedef float v8f __attribute__((ext_vector_type(8)));

// ---------------- zero scratch ----------------
__global__ void k_zero(float* p, size_t n) {
    size_t i = (size_t)blockIdx.x * blockDim.x + threadIdx.x;
    size_t stride = (size_t)gridDim.x * blockDim.x;
    for (; i < n; i += stride) p[i] = 0.0f;
}

// ---------------- fused embedding gather + msg linear ----------------
// h[node, :] = msg_W @ x[node] + msg_b,  x = [price | cat | sub | elem | brand | item]
__global__ __launch_bounds__(128) void k_embed_msg(
    const int* __restrict__ cat, const int* __restrict__ sub,
    const int* __restrict__ elem, const int* __restrict__ brand,
    const int* __restrict__ pid, const float* __restrict__ price,
    const float* __restrict__ ecat, const float* __restrict__ esub,
    const float* __restrict__ eelem, const float* __restrict__ ebrand,
    const float* __restrict__ eitem,
    const float* __restrict__ W, const float* __restrict__ b,
    float* __restrict__ h)
{
    __shared__ float xs[IN_DIM];
    int node = blockIdx.x;
    int t = threadIdx.x;
    for (int j = t; j < IN_DIM; j += 128) {
        float v;
        if (j == 0) {
            v = price[node];
        } else {
            int jj = j - 1, r = jj >> 6, o = jj & 63;
            const float* tab; int idx;
            switch (r) {
                case 0:  tab = ecat;   idx = cat[node];   break;
                case 1:  tab = esub;   idx = sub[node];   break;
                case 2:  tab = eelem;  idx = elem[node];  break;
                case 3:  tab = ebrand; idx = brand[node]; break;
                default: tab = eitem;  idx = pid[node];   break;
            }
            v = tab[(size_t)idx * EMB + o];
        }
        xs[j] = v;
    }
    __syncthreads();
    if (t < HID) {
        float acc = b[t];
        const float* wr = W + (size_t)t * IN_DIM;
        #pragma unroll 4
        for (int k = 0; k < IN_DIM; ++k) acc = fmaf(xs[k], wr[k], acc);
        h[(size_t)node * HID + t] = acc;
    }
}

// ---------------- edge scatter: sums[dst] += h[src], counts[dst] += 1 ----------------
__global__ void k_edge_scatter(const int* __restrict__ ei,
                               const float* __restrict__ h,
                               float* __restrict__ sums,
                               float* __restrict__ cnt)
{
    int e = blockIdx.x * blockDim.x + threadIdx.x;
    if (e >= N_EDGES) return;
    int src = ei[e];
    int dst = ei[N_EDGES + e];
    atomicAdd(&cnt[dst], 1.0f);
    const float* hs = h + (size_t)src * HID;
    float* sd = sums + (size_t)dst * HID;
    for (int d = 0; d < HID; ++d) atomicAdd(&sd[d], hs[d]);
}

// ---------------- fused msgs-mean + GRU cell via f32 WMMA ----------------
// per block: 16 nodes.  gi = msgs @ Wih^T + bih ; gh = h @ Whh^T + bhh ; GRU combine.
__global__ __launch_bounds__(256) void k_gru(
    const float* __restrict__ sums, const float* __restrict__ cnt,
    const float* __restrict__ h,
    const float* __restrict__ Wih, const float* __restrict__ Whh,
    const float* __restrict__ bih, const float* __restrict__ bhh,
    float* __restrict__ h_new)
{
    __shared__ float ms[16 * HID];
    __shared__ float hs[16 * HID];
    __shared__ float gi[16 * 300];
    __shared__ float gh[16 * 300];

    int tile = blockIdx.x;
    int t = threadIdx.x;

    for (int idx = t; idx < 16 * HID; idx += 256) {
        int m = idx / HID, d = idx % HID;
        int node = tile * 16 + m;
        float c = cnt[node];
        c = c > 1.0f ? c : 1.0f;
        ms[idx] = sums[(size_t)node * HID + d] / c;
        hs[idx] = h[(size_t)node * HID + d];
    }
    __syncthreads();

    int wave = t >> 5, lane = t & 31;
    int laneM = lane & 15;          // A row / B col / D col
    int kb = (lane >> 4) << 1;      // K sub-offset: lanes 16-31 hold K+2,K+3
    int rowbase = (lane >> 4) << 3; // D rows: r + 8*(lane/16)

    for (int nt = wave; nt < NTILES_GATE; nt += 8) {
        int col = nt * 16 + laneM;            // gate index 0..303
        bool okc = col < 300;
        int colc = okc ? col : 299;           // clamp: loads stay straight-line,
                                              // garbage lanes discarded at write
        const float* wih_r = Wih + (size_t)colc * HID + kb;
        const float* whh_r = Whh + (size_t)colc * HID + kb;
        const float* ms_r  = ms + laneM * HID + kb;
        const float* hs_r  = hs + laneM * HID + kb;

        v8f c1 = {0.f,0.f,0.f,0.f,0.f,0.f,0.f,0.f};
        v8f c2 = {0.f,0.f,0.f,0.f,0.f,0.f,0.f,0.f};
        #pragma unroll 5
        for (int k0 = 0; k0 < HID; k0 += 4) {
            v2f a1, a2, b1, b2;
            a1.x = ms_r[k0];  a1.y = ms_r[k0 + 1];
            a2.x = hs_r[k0];  a2.y = hs_r[k0 + 1];
            b1.x = wih_r[k0]; b1.y = wih_r[k0 + 1];
            b2.x = whh_r[k0]; b2.y = whh_r[k0 + 1];
            c1 = __builtin_amdgcn_wmma_f32_16x16x4_f32(false, a1, false, b1,
                                                       (short)0, c1, false, false);
            c2 = __builtin_amdgcn_wmma_f32_16x16x4_f32(false, a2, false, b2,
                                                       (short)0, c2, false, false);
        }
        if (okc) {
            float bi = bih[col], bh = bhh[col];
            #pragma unroll
            for (int r = 0; r < 8; ++r) {
                gi[(rowbase + r) * 300 + col] = c1[r] + bi;
                gh[(rowbase + r) * 300 + col] = c2[r] + bh;
            }
        }
    }
    __syncthreads();

    for (int idx = t; idx < 16 * HID; idx += 256) {
        int m = idx / HID, d = idx % HID;
        float i_r = gi[m * 300 + d],       h_r = gh[m * 300 + d];
        float i_z = gi[m * 300 + 100 + d], h_z = gh[m * 300 + 100 + d];
        float i_n = gi[m * 300 + 200 + d], h_n = gh[m * 300 + 200 + d];
        float r = 1.f / (1.f + __expf(-(i_r + h_r)));
        float z = 1.f / (1.f + __expf(-(i_z + h_z)));
        float n = tanhf(i_n + r * h_n);
        float hv = (1.f - z) * n + z * hs[idx];
        h_new[(size_t)(tile * 16 + m) * HID + d] = hv;  // overwrites consumed sums
    }
}

// ---------------- global mean pool ----------------
__global__ void k_pool_sum(const float* __restrict__ hn, const int* __restrict__ batch,
                           float* __restrict__ g, float* __restrict__ gcnt)
{
    int node = blockIdx.x * blockDim.x + threadIdx.x;
    if (node >= N_NODES) return;
    int b = batch[node];
    atomicAdd(&gcnt[b], 1.0f);
    const float* hr = hn + (size_t)node * HID;
    float* gr = g + (size_t)b * HID;
    for (int d = 0; d < HID; ++d) atomicAdd(&gr[d], hr[d]);
}

__global__ void k_pool_div(float* __restrict__ g, const float* __restrict__ gcnt)
{
    int i = blockIdx.x * blockDim.x + threadIdx.x;
    if (i >= N_GRAPHS * HID) return;
    float c = gcnt[i / HID];
    c = c > 1.0f ? c : 1.0f;
    g[i] /= c;
}

// ---------------- scores = g @ fc_W^T + fc_b  (f32 WMMA, output-stream bound) ------
// A tile (g, 16x100) staged once per block in LDS and shared by 4 waves;
// result streamed out with non-temporal stores (written once, never re-read).
__global__ __launch_bounds__(128) void k_fc(const float* __restrict__ g,
                                            const float* __restrict__ W,
                                            const float* __restrict__ bias,
                                            float* __restrict__ out)
{
    __shared__ float gs[16 * HID];
    int m0 = blockIdx.y * 16;
    for (int idx = threadIdx.x; idx < 16 * HID; idx += 128)
        gs[idx] = g[(size_t)(m0 + idx / HID) * HID + (idx % HID)];
    __syncthreads();

    int wave = threadIdx.x >> 5, lane = threadIdx.x & 31;
    int nt = blockIdx.x * 4 + wave;       // N tile (of 3125)
    if (nt >= NUM_ITEMS / 16) return;     // wave-uniform exit (no barriers follow)

    int laneM = lane & 15;
    int kb = (lane >> 4) << 1;
    int rowbase = (lane >> 4) << 3;
    int col = nt * 16 + laneM;

    const float* ga = gs + laneM * HID + kb;
    const float* wb = W + (size_t)col * HID + kb;

    v8f c = {0.f,0.f,0.f,0.f,0.f,0.f,0.f,0.f};
    #pragma unroll
    for (int k0 = 0; k0 < HID; k0 += 4) {
        v2f a, b;
        a.x = ga[k0]; a.y = ga[k0 + 1];
        b.x = wb[k0]; b.y = wb[k0 + 1];
        c = __builtin_amdgcn_wmma_f32_16x16x4_f32(false, a, false, b,
                                                  (short)0, c, false, false);
    }
    float bv = bias[col];
    #pragma unroll
    for (int r = 0; r < 8; ++r) {
        float v = c[r] + bv;
        __builtin_nontemporal_store(v, &out[(size_t)(m0 + rowbase + r) * NUM_ITEMS + col]);
    }
}

// ---------------- launch ----------------
extern "C" void kernel_launch(void* const* d_in, const int* in_sizes, int n_in,
                              void* d_out, int out_size, void* d_ws, size_t ws_size,
                              hipStream_t stream)
{
    const int*   cat    = (const int*)  d_in[0];
    const int*   sub    = (const int*)  d_in[1];
    const int*   elem   = (const int*)  d_in[2];
    const int*   brand  = (const int*)  d_in[3];
    const int*   pid    = (const int*)  d_in[4];
    const float* price  = (const float*)d_in[5];
    const int*   ei     = (const int*)  d_in[6];
    const int*   batch  = (const int*)  d_in[7];
    const float* ecat   = (const float*)d_in[8];
    const float* esub   = (const float*)d_in[9];
    const float* eelem  = (const float*)d_in[10];
    const float* ebrand = (const float*)d_in[11];
    const float* eitem  = (const float*)d_in[12];
    const float* msgW   = (const float*)d_in[13];
    const float* msgb   = (const float*)d_in[14];
    const float* Wih    = (const float*)d_in[15];
    const float* Whh    = (const float*)d_in[16];
    const float* bih    = (const float*)d_in[17];
    const float* bhh    = (const float*)d_in[18];
    const float* fcW    = (const float*)d_in[19];
    const float* fcb    = (const float*)d_in[20];
    float* out = (float*)d_out;

    // workspace layout (floats)
    float* h    = (float*)d_ws;                      // N_NODES*HID = 10,000,000
    float* sums = h    + (size_t)N_NODES * HID;      // 10,000,000 (msgs, then h_new)
    float* cnts = sums + (size_t)N_NODES * HID;      // 100,000
    float* g    = cnts + N_NODES;                    // 409,600
    float* gcnt = g    + (size_t)N_GRAPHS * HID;     // 4,096

    size_t nzero = (size_t)N_NODES * HID + N_NODES + (size_t)N_GRAPHS * HID + N_GRAPHS;
    k_zero<<<2048, 256, 0, stream>>>(sums, nzero);

    k_embed_msg<<<N_NODES, 128, 0, stream>>>(cat, sub, elem, brand, pid, price,
                                             ecat, esub, eelem, ebrand, eitem,
                                             msgW, msgb, h);

    k_edge_scatter<<<(N_EDGES + 255) / 256, 256, 0, stream>>>(ei, h, sums, cnts);

    k_gru<<<N_NODES / 16, 256, 0, stream>>>(sums, cnts, h, Wih, Whh, bih, bhh, sums);

    k_pool_sum<<<(N_NODES + 255) / 256, 256, 0, stream>>>(sums, batch, g, gcnt);
    k_pool_div<<<(N_GRAPHS * HID + 255) / 256, 256, 0, stream>>>(g, gcnt);

    dim3 fc_grid((NUM_ITEMS / 16 + 3) / 4, N_GRAPHS / 16);
    k_fc<<<fc_grid, 128, 0, stream>>>(g, fcW, fcb, out);
}